// GATEncoder_31963146617271
// MI455X (gfx1250) — compile-verified
//
#include <hip/hip_runtime.h>
#include <math.h>

// GAT encoder for MI455X (gfx1250, wave32). Dense transforms use
// V_WMMA_F32_16X16X4_F32 (full fp32). Edge passes are atomic-based
// segment ops that stay resident in the 192MB L2.

typedef __attribute__((ext_vector_type(2))) float v2f;
typedef __attribute__((ext_vector_type(8))) float v8f;

#define NEG_SLOPE 0.2f

static __device__ __forceinline__ unsigned f2ord(float f) {
    unsigned b = __float_as_uint(f);
    return (b & 0x80000000u) ? ~b : (b | 0x80000000u);
}
static __device__ __forceinline__ float ord2f(unsigned o) {
    unsigned b = (o & 0x80000000u) ? (o ^ 0x80000000u) : ~o;
    return __uint_as_float(b);
}

__global__ void zero_kernel(float* __restrict__ p, long long n) {
    long long i = (long long)blockIdx.x * blockDim.x + threadIdx.x;
    long long stride = (long long)gridDim.x * blockDim.x;
    for (; i < n; i += stride) p[i] = 0.0f;
}

// ---- Layer 1 GEMM: h1[N,64] = x[N,4] @ W1[4,64], one wave per 16-row tile.
// K = 4 exactly matches one V_WMMA_F32_16X16X4_F32.
__global__ __launch_bounds__(32) void gemm1_wmma(const float* __restrict__ x,
                                                 const float* __restrict__ W1,
                                                 float* __restrict__ h1) {
    const int lane = threadIdx.x;            // 0..31
    const int row0 = blockIdx.x * 16;
    const int m  = lane & 15;
    const int hi = lane >> 4;                // 0 or 1
    const int kb = hi * 2;                   // K base for this half-wave
    v2f a;
    a.x = x[(row0 + m) * 4 + kb + 0];
    a.y = x[(row0 + m) * 4 + kb + 1];
#pragma unroll
    for (int t = 0; t < 4; ++t) {            // 4 N-tiles of 16 cols => 64 cols
        const int n = t * 16 + m;
        v2f b;
        b.x = W1[(kb + 0) * 64 + n];
        b.y = W1[(kb + 1) * 64 + n];
        v8f c = {};
        c = __builtin_amdgcn_wmma_f32_16x16x4_f32(false, a, false, b,
                                                  (short)0, c, false, false);
#pragma unroll
        for (int j = 0; j < 8; ++j)          // C: row = j + 8*hi, col = n
            h1[(size_t)(row0 + j + hi * 8) * 64 + n] = c[j];
    }
}

// ---- Layer 2 GEMM: h2[N,8] = x1[N,64] @ W2[64,8]; chain 16 K=4 WMMAs, pad N->16.
__global__ __launch_bounds__(32) void gemm2_wmma(const float* __restrict__ x1,
                                                 const float* __restrict__ W2,
                                                 float* __restrict__ h2) {
    const int lane = threadIdx.x;
    const int row0 = blockIdx.x * 16;
    const int m  = lane & 15;
    const int hi = lane >> 4;
    v8f c = {};
#pragma unroll
    for (int kk = 0; kk < 16; ++kk) {
        const int kb = kk * 4 + hi * 2;
        v2f a, b;
        a.x = x1[(size_t)(row0 + m) * 64 + kb + 0];
        a.y = x1[(size_t)(row0 + m) * 64 + kb + 1];
        b.x = (m < 8) ? W2[(kb + 0) * 8 + m] : 0.0f;
        b.y = (m < 8) ? W2[(kb + 1) * 8 + m] : 0.0f;
        c = __builtin_amdgcn_wmma_f32_16x16x4_f32(false, a, false, b,
                                                  (short)0, c, false, false);
    }
    if (m < 8) {
#pragma unroll
        for (int j = 0; j < 8; ++j)
            h2[(size_t)(row0 + j + hi * 8) * 8 + m] = c[j];
    }
}

// ---- Attention coefficients, layer 1: a_s/a_d[N,4] = sum_c h1[n,h,c]*att[h,c]
__global__ void att1_kernel(const float* __restrict__ h1,
                            const float* __restrict__ att_s,
                            const float* __restrict__ att_d,
                            float* __restrict__ as, float* __restrict__ ad, int n4) {
    int t = blockIdx.x * blockDim.x + threadIdx.x;
    if (t >= n4) return;
    int node = t >> 2, head = t & 3;
    const float* hp = h1 + (size_t)node * 64 + head * 16;
    const float* sp = att_s + head * 16;
    const float* dp = att_d + head * 16;
    float s = 0.f, d = 0.f;
#pragma unroll
    for (int c = 0; c < 16; ++c) { s += hp[c] * sp[c]; d += hp[c] * dp[c]; }
    as[t] = s; ad[t] = d;
}

__global__ void att2_kernel(const float* __restrict__ h2,
                            const float* __restrict__ att_s,
                            const float* __restrict__ att_d,
                            float* __restrict__ as, float* __restrict__ ad, int n) {
    int t = blockIdx.x * blockDim.x + threadIdx.x;
    if (t >= n) return;
    const float* hp = h2 + (size_t)t * 8;
    float s = 0.f, d = 0.f;
#pragma unroll
    for (int c = 0; c < 8; ++c) { s += hp[c] * att_s[c]; d += hp[c] * att_d[c]; }
    as[t] = s; ad[t] = d;
}

static __device__ __forceinline__ void edge_sd(const int* __restrict__ ei, int E,
                                               int e, int& s, int& d) {
    if (e < E) { s = ei[e]; d = ei[E + e]; }
    else       { s = e - E; d = e - E; }     // appended self-loops
}

// ---- Edge pass A: segment max of leaky_relu(a_s[src]+a_d[dst]) via ordered-uint atomicMax
__global__ void edge_max1(const int* __restrict__ ei, int E, int etot4,
                          const float* __restrict__ as, const float* __restrict__ ad,
                          unsigned* __restrict__ m1) {
    int t = blockIdx.x * blockDim.x + threadIdx.x;
    if (t >= etot4) return;
    int e = t >> 2, head = t & 3, s, d;
    edge_sd(ei, E, e, s, d);
    float v = as[s * 4 + head] + ad[d * 4 + head];
    v = (v > 0.f) ? v : NEG_SLOPE * v;
    atomicMax(&m1[d * 4 + head], f2ord(v));
}

// ---- Edge pass B: w = exp(e - m); s[dst]+=w; agg[dst]+= w*h1[src] (fused, unnormalized)
__global__ void edge_msg1(const int* __restrict__ ei, int E, int etot4,
                          const float* __restrict__ as, const float* __restrict__ ad,
                          const unsigned* __restrict__ m1,
                          const float* __restrict__ h1,
                          float* __restrict__ s1, float* __restrict__ agg1) {
    int t = blockIdx.x * blockDim.x + threadIdx.x;
    if (t >= etot4) return;
    int e = t >> 2, head = t & 3, s, d;
    edge_sd(ei, E, e, s, d);
    float v = as[s * 4 + head] + ad[d * 4 + head];
    v = (v > 0.f) ? v : NEG_SLOPE * v;
    float w = expf(v - ord2f(m1[d * 4 + head]));
    atomicAdd(&s1[d * 4 + head], w);
    const float4* hp = (const float4*)(h1 + (size_t)s * 64 + head * 16);
    float* ap = agg1 + (size_t)d * 64 + head * 16;
#pragma unroll
    for (int g = 0; g < 4; ++g) {
        float4 hv = hp[g];
        atomicAdd(&ap[g * 4 + 0], w * hv.x);
        atomicAdd(&ap[g * 4 + 1], w * hv.y);
        atomicAdd(&ap[g * 4 + 2], w * hv.z);
        atomicAdd(&ap[g * 4 + 3], w * hv.w);
    }
}

__global__ void edge_max2(const int* __restrict__ ei, int E, int etot,
                          const float* __restrict__ as, const float* __restrict__ ad,
                          unsigned* __restrict__ m2) {
    int e = blockIdx.x * blockDim.x + threadIdx.x;
    if (e >= etot) return;
    int s, d; edge_sd(ei, E, e, s, d);
    float v = as[s] + ad[d];
    v = (v > 0.f) ? v : NEG_SLOPE * v;
    atomicMax(&m2[d], f2ord(v));
}

__global__ void edge_msg2(const int* __restrict__ ei, int E, int etot,
                          const float* __restrict__ as, const float* __restrict__ ad,
                          const unsigned* __restrict__ m2,
                          const float* __restrict__ h2,
                          float* __restrict__ s2, float* __restrict__ agg2) {
    int e = blockIdx.x * blockDim.x + threadIdx.x;
    if (e >= etot) return;
    int s, d; edge_sd(ei, E, e, s, d);
    float v = as[s] + ad[d];
    v = (v > 0.f) ? v : NEG_SLOPE * v;
    float w = expf(v - ord2f(m2[d]));
    atomicAdd(&s2[d], w);
    const float4* hp = (const float4*)(h2 + (size_t)s * 8);
    float* ap = agg2 + (size_t)d * 8;
#pragma unroll
    for (int g = 0; g < 2; ++g) {
        float4 hv = hp[g];
        atomicAdd(&ap[g * 4 + 0], w * hv.x);
        atomicAdd(&ap[g * 4 + 1], w * hv.y);
        atomicAdd(&ap[g * 4 + 2], w * hv.z);
        atomicAdd(&ap[g * 4 + 3], w * hv.w);
    }
}

// ---- Node finalize, layer 1: x1 = elu(agg/s + b1)
__global__ void node1_kernel(const float* __restrict__ agg1, const float* __restrict__ s1,
                             const float* __restrict__ b1, float* __restrict__ x1, int n64) {
    int t = blockIdx.x * blockDim.x + threadIdx.x;
    if (t >= n64) return;
    int node = t >> 6, j = t & 63, head = j >> 4;
    float v = agg1[t] / s1[node * 4 + head] + b1[j];
    x1[t] = (v > 0.f) ? v : (expf(v) - 1.0f);
}

// ---- Node finalize, layer 2: out = agg/s + b2
__global__ void node2_kernel(const float* __restrict__ agg2, const float* __restrict__ s2,
                             const float* __restrict__ b2, float* __restrict__ out, int n8) {
    int t = blockIdx.x * blockDim.x + threadIdx.x;
    if (t >= n8) return;
    int node = t >> 3, c = t & 7;
    out[t] = agg2[t] / s2[node] + b2[c];
}

static inline int cdiv(long long a, int b) { return (int)((a + b - 1) / b); }

extern "C" void kernel_launch(void* const* d_in, const int* in_sizes, int n_in,
                              void* d_out, int out_size, void* d_ws, size_t ws_size,
                              hipStream_t stream) {
    const float* x    = (const float*)d_in[0];
    const float* W1   = (const float*)d_in[1];
    const float* aS1  = (const float*)d_in[2];
    const float* aD1  = (const float*)d_in[3];
    const float* b1   = (const float*)d_in[4];
    const float* W2   = (const float*)d_in[5];
    const float* aS2  = (const float*)d_in[6];
    const float* aD2  = (const float*)d_in[7];
    const float* b2   = (const float*)d_in[8];
    const int*   ei   = (const int*)d_in[9];   // [2,E] flat: row0=src, row1=dst

    const int N = in_sizes[0] / 4;             // 50000 (multiple of 16)
    const int E = in_sizes[9] / 2;             // 800000
    const int Etot = E + N;                    // with self-loops

    // Workspace layout (floats)
    float* ws   = (float*)d_ws;
    float* h1   = ws;                              // N*64
    float* x1   = h1  + (size_t)N * 64;            // N*64
    float* h2   = x1  + (size_t)N * 64;            // N*8
    float* as1  = h2  + (size_t)N * 8;             // N*4
    float* ad1  = as1 + (size_t)N * 4;             // N*4
    float* as2  = ad1 + (size_t)N * 4;             // N
    float* ad2  = as2 + (size_t)N;                 // N
    float* zb   = ad2 + (size_t)N;                 // zeroed region start
    unsigned* m1   = (unsigned*)zb;                // N*4
    float*    s1   = zb + (size_t)N * 4;           // N*4
    float*    agg1 = zb + (size_t)N * 8;           // N*64
    unsigned* m2   = (unsigned*)(zb + (size_t)N * 72); // N
    float*    s2   = zb + (size_t)N * 73;          // N
    float*    agg2 = zb + (size_t)N * 74;          // N*8
    const long long nz = (long long)N * 82;

    zero_kernel<<<512, 256, 0, stream>>>(zb, nz);

    // ---- layer 1
    gemm1_wmma<<<N / 16, 32, 0, stream>>>(x, W1, h1);
    att1_kernel<<<cdiv((long long)N * 4, 256), 256, 0, stream>>>(h1, aS1, aD1, as1, ad1, N * 4);
    edge_max1<<<cdiv((long long)Etot * 4, 256), 256, 0, stream>>>(ei, E, Etot * 4, as1, ad1, m1);
    edge_msg1<<<cdiv((long long)Etot * 4, 256), 256, 0, stream>>>(ei, E, Etot * 4, as1, ad1, m1, h1, s1, agg1);
    node1_kernel<<<cdiv((long long)N * 64, 256), 256, 0, stream>>>(agg1, s1, b1, x1, N * 64);

    // ---- layer 2
    gemm2_wmma<<<N / 16, 32, 0, stream>>>(x1, W2, h2);
    att2_kernel<<<cdiv(N, 256), 256, 0, stream>>>(h2, aS2, aD2, as2, ad2, N);
    edge_max2<<<cdiv(Etot, 256), 256, 0, stream>>>(ei, E, Etot, as2, ad2, m2);
    edge_msg2<<<cdiv(Etot, 256), 256, 0, stream>>>(ei, E, Etot, as2, ad2, m2, h2, s2, agg2);
    node2_kernel<<<cdiv((long long)N * 8, 256), 256, 0, stream>>>(agg2, s2, b2, (float*)d_out, N * 8);
}